// SpatialAttention_41334765257060
// MI455X (gfx1250) — compile-verified
//
#include <hip/hip_runtime.h>
#include <hip/hip_bf16.h>

#define NEG_SLOPE 0.001f
#define HH 160
#define WW 160
#define HW (HH*WW)        // 25600
#define BATCH 16
#define CCH 256
#define NPIX (BATCH*HW)   // 409600

typedef float v2f __attribute__((ext_vector_type(2)));
typedef float v8f __attribute__((ext_vector_type(8)));

__device__ __forceinline__ float lrelu(float v){ return v > 0.f ? v : NEG_SLOPE * v; }

// ---------------------------------------------------------------------------
// K1: one fused pass over x (419 MB): channel mean, channel max, Bm = lrelu(w_ab . x)
// float4 (b128) loads: each thread owns 4 consecutive pixels, loops 256 channels.
// ---------------------------------------------------------------------------
__global__ void k1_reduce(const float* __restrict__ x, const float* __restrict__ w_ab,
                          float* __restrict__ avg, float* __restrict__ mx,
                          float* __restrict__ bm){
    int gid = blockIdx.x * blockDim.x + threadIdx.x;     // 0 .. NPIX/4-1
    int b = gid / (HW/4);
    int p = (gid % (HW/4)) * 4;                          // 16B-aligned pixel group
    const float4* xb4 = reinterpret_cast<const float4*>(x + (size_t)b * CCH * HW + p);
    float4 s  = make_float4(0.f, 0.f, 0.f, 0.f);
    float4 ac = make_float4(0.f, 0.f, 0.f, 0.f);
    float4 m  = make_float4(-3.402823466e38f, -3.402823466e38f,
                            -3.402823466e38f, -3.402823466e38f);
    for (int c = 0; c < CCH; ++c){
        float4 v = xb4[(size_t)c * (HW/4)];
        float wc = w_ab[c];
        s.x += v.x; s.y += v.y; s.z += v.z; s.w += v.w;
        m.x = fmaxf(m.x, v.x); m.y = fmaxf(m.y, v.y);
        m.z = fmaxf(m.z, v.z); m.w = fmaxf(m.w, v.w);
        ac.x = fmaf(wc, v.x, ac.x); ac.y = fmaf(wc, v.y, ac.y);
        ac.z = fmaf(wc, v.z, ac.z); ac.w = fmaf(wc, v.w, ac.w);
    }
    size_t o = (size_t)b * HW + p;
    const float inv = 1.f / CCH;
    *reinterpret_cast<float4*>(avg + o) = make_float4(s.x*inv, s.y*inv, s.z*inv, s.w*inv);
    *reinterpret_cast<float4*>(mx  + o) = m;
    *reinterpret_cast<float4*>(bm  + o) = make_float4(lrelu(ac.x), lrelu(ac.y),
                                                      lrelu(ac.z), lrelu(ac.w));
}

// ---------------------------------------------------------------------------
// K2: the 1/3/5/7 spatial convs on the 2-channel (avg,max) map, fused with
// Cm = lrelu(w_fc . F). Input planes are tiny (3.3 MB) and L2-resident.
// ---------------------------------------------------------------------------
__global__ void k2_convs(const float* __restrict__ avg, const float* __restrict__ mx,
                         const float* __restrict__ w1, const float* __restrict__ w3,
                         const float* __restrict__ w5, const float* __restrict__ w7,
                         const float* __restrict__ w_fc,
                         float* __restrict__ F, float* __restrict__ Cm){
    __shared__ float wl[172];           // [0]=w1(2) [2]=w3(18) [20]=w5(50) [70]=w7(98) [168]=w_fc(4)
    int tid = threadIdx.x;
    if (tid < 2)  wl[tid]      = w1[tid];
    if (tid < 18) wl[2 + tid]  = w3[tid];
    if (tid < 50) wl[20 + tid] = w5[tid];
    if (tid < 98) wl[70 + tid] = w7[tid];
    if (tid < 4)  wl[168 + tid]= w_fc[tid];
    __syncthreads();

    int blk = blockIdx.x;               // 16 batches x 100 tiles of 16x16
    int b = blk / 100, t = blk % 100;
    int h = (t / 10) * 16 + tid / 16;
    int w = (t % 10) * 16 + tid % 16;
    const float* A = avg + (size_t)b * HW;
    const float* M = mx  + (size_t)b * HW;

    float s1 = wl[0] * A[h*WW + w] + wl[1] * M[h*WW + w];
    float s3 = 0.f, s5 = 0.f, s7 = 0.f;
    for (int dy = -3; dy <= 3; ++dy){
        int hh = h + dy;
        for (int dx = -3; dx <= 3; ++dx){
            int ww2 = w + dx;
            float a = 0.f, m2 = 0.f;
            if (hh >= 0 && hh < HH && ww2 >= 0 && ww2 < WW){
                a = A[hh*WW + ww2]; m2 = M[hh*WW + ww2];
            }
            s7 += wl[70 + (dy+3)*7 + (dx+3)] * a + wl[70+49 + (dy+3)*7 + (dx+3)] * m2;
            if (dy >= -2 && dy <= 2 && dx >= -2 && dx <= 2)
                s5 += wl[20 + (dy+2)*5 + (dx+2)] * a + wl[20+25 + (dy+2)*5 + (dx+2)] * m2;
            if (dy >= -1 && dy <= 1 && dx >= -1 && dx <= 1)
                s3 += wl[2 + (dy+1)*3 + (dx+1)] * a + wl[2+9 + (dy+1)*3 + (dx+1)] * m2;
        }
    }
    s1 = lrelu(s1); s3 = lrelu(s3); s5 = lrelu(s5); s7 = lrelu(s7);
    size_t base = ((size_t)b * 4) * HW + h*WW + w;
    F[base]        = s1;
    F[base +   HW] = s3;
    F[base + 2*HW] = s5;
    F[base + 3*HW] = s7;
    Cm[(size_t)b * HW + h*WW + w] =
        lrelu(wl[168]*s1 + wl[169]*s3 + wl[170]*s5 + wl[171]*s7);
}

// ---------------------------------------------------------------------------
// K3: M[b,h,k] = sum_w Bm[b,h,w] * Cm[b, w*160 + k]  (reshape(B,W,H) == same memory
// since H==W), then softmax over k. One block per (b,h) row; 131 MFLOP total.
// ---------------------------------------------------------------------------
__global__ void k3_attn(const float* __restrict__ Bm, const float* __restrict__ Cm,
                        float* __restrict__ S){
    __shared__ float red[160];
    int bh = blockIdx.x;                 // 16 * 160
    int b = bh / HH, h = bh % HH;
    int k = threadIdx.x;                 // 0..159
    const float* brow = Bm + (size_t)b * HW + (size_t)h * WW;
    const float* cmat = Cm + (size_t)b * HW;
    float acc = 0.f;
    for (int w = 0; w < WW; ++w)
        acc = fmaf(brow[w], cmat[w*HH + k], acc);

    red[k] = acc; __syncthreads();
    for (int off = 128; off > 0; off >>= 1){
        if (k < off && k + off < 160) red[k] = fmaxf(red[k], red[k + off]);
        __syncthreads();
    }
    float mval = red[0]; __syncthreads();
    float e = expf(acc - mval);
    red[k] = e; __syncthreads();
    for (int off = 128; off > 0; off >>= 1){
        if (k < off && k + off < 160) red[k] += red[k + off];
        __syncthreads();
    }
    S[(size_t)bh * 160 + k] = e / red[0];
}

// ---------------------------------------------------------------------------
// K4 (big GEMM, 53.7 GFLOP): G[b,o,p] = sum_c w_e[o,c] * lrelu(w_fd[c,:] . F[b,:,p])
// D is never materialized: the 256 x 32 D-tile is built in LDS from the 4-channel
// F tile, then each of 8 waves runs a 16x16 fp32 WMMA tile (K=256 in 64 steps).
// G is written into d_out (reused as scratch; fully overwritten by K5).
// ---------------------------------------------------------------------------
#define K4_PS 33   // padded LDS row stride for the D tile
__global__ void k4_gemm_we(const float* __restrict__ Ff, const float* __restrict__ w_fd,
                           const float* __restrict__ w_e, float* __restrict__ G){
    __shared__ float Dlds[CCH * K4_PS];     // 256 x 32 (padded) = 33.8 KB
    __shared__ float wfd[CCH * 4];
    __shared__ float Flds[4 * 32];
    int tid = threadIdx.x;                  // 256 threads = 8 waves
    int b = blockIdx.z;                     // 16
    int otile = blockIdx.y;                 // 4  -> o block of 64
    int p0 = blockIdx.x * 32;               // 800 -> pixel block of 32

    if (tid < 128){
        int j = tid / 32, pl = tid % 32;
        Flds[j*32 + pl] = Ff[((size_t)b*4 + j) * HW + p0 + pl];
    }
    for (int i = tid; i < CCH*4; i += 256) wfd[i] = w_fd[i];
    __syncthreads();

    for (int i = tid; i < CCH*32; i += 256){
        int c = i >> 5, pl = i & 31;
        float d = wfd[c*4+0]*Flds[pl]      + wfd[c*4+1]*Flds[32+pl]
                + wfd[c*4+2]*Flds[64+pl]   + wfd[c*4+3]*Flds[96+pl];
        Dlds[c*K4_PS + pl] = lrelu(d);
    }
    __syncthreads();

    int wave = tid >> 5, lane = tid & 31;
    int ow = wave & 3, pw = wave >> 2;      // 4 o-subtiles x 2 p-subtiles
    int m = lane & 15, half = lane >> 4, n = lane & 15;
    int orow = otile*64 + ow*16 + m;
    int pl = pw*16 + n;
    const float* we_row = w_e + (size_t)orow * CCH;   // 256 KB matrix, L2-hot

    v8f acc = {};
    for (int kk = 0; kk < 64; ++kk){
        int c0 = kk*4 + 2*half;
        v2f a, bf;
        a.x  = we_row[c0];
        a.y  = we_row[c0 + 1];
        bf.x = Dlds[c0 * K4_PS + pl];
        bf.y = Dlds[(c0 + 1) * K4_PS + pl];
        acc = __builtin_amdgcn_wmma_f32_16x16x4_f32(false, a, false, bf,
                                                    (short)0, acc, false, false);
    }
    size_t gbase = ((size_t)b * CCH) * HW;
    for (int j = 0; j < 8; ++j){
        int o = otile*64 + ow*16 + j + 8*half;
        G[gbase + (size_t)o * HW + p0 + pl] = acc[j];
    }
}

// ---------------------------------------------------------------------------
// K5 (33.6 GFLOP): out[b,o,h,w] = sum_k S[b,h,k] * G[b,o,k,w] + x[b,o,h,w].
// In-place over d_out: each block owns one (b,o) 160x160 G-plane, stages it
// entirely in LDS (100 KB; CDNA5 WGP has 320 KB) with b128 copies, then 10
// waves compute the 100 WMMA tiles (K=160 in 40 fp32 wmma steps) and
// overwrite the plane. Reads complete before any write -> in-place safe.
// ---------------------------------------------------------------------------
__global__ void k5_sgemm_out(const float* __restrict__ S, const float* __restrict__ x,
                             float* __restrict__ outG){
    __shared__ float Gl[HW];               // 160*160 floats = 100 KB
    int tid = threadIdx.x;                 // 320 threads = 10 waves
    int b = blockIdx.y, o = blockIdx.x;
    size_t plane = ((size_t)b * CCH + o) * HW;

    {   // stage full plane with float4 (b128) loads: 20 per thread
        const float4* src4 = reinterpret_cast<const float4*>(outG + plane);
        float4* Gl4 = reinterpret_cast<float4*>(Gl);
        for (int i = tid; i < HW/4; i += 320) Gl4[i] = src4[i];
    }
    __syncthreads();                       // full plane in LDS before any write

    int wave = tid >> 5, lane = tid & 31;
    int m = lane & 15, half = lane >> 4, n = lane & 15;
    const float* Sb = S + (size_t)b * HW;

    for (int t = wave; t < 100; t += 10){  // 10x10 tiles of 16x16
        int h0 = (t / 10) * 16, w0 = (t % 10) * 16;
        const float* Srow = Sb + (size_t)(h0 + m) * 160;
        v8f acc = {};
        for (int kk = 0; kk < 40; ++kk){
            int k0 = kk*4 + 2*half;
            v2f a, bf;
            a.x  = Srow[k0];
            a.y  = Srow[k0 + 1];
            bf.x = Gl[k0 * WW + w0 + n];
            bf.y = Gl[(k0 + 1) * WW + w0 + n];
            acc = __builtin_amdgcn_wmma_f32_16x16x4_f32(false, a, false, bf,
                                                        (short)0, acc, false, false);
        }
        for (int j = 0; j < 8; ++j){
            int h = h0 + j + 8*half;
            size_t idx2 = plane + (size_t)h * WW + w0 + n;
            outG[idx2] = acc[j] + x[idx2];
        }
    }
}

// ---------------------------------------------------------------------------
extern "C" void kernel_launch(void* const* d_in, const int* in_sizes, int n_in,
                              void* d_out, int out_size, void* d_ws, size_t ws_size,
                              hipStream_t stream){
    const float* x    = (const float*)d_in[0];
    const float* w1   = (const float*)d_in[1];
    const float* w3   = (const float*)d_in[2];
    const float* w5   = (const float*)d_in[3];
    const float* w7   = (const float*)d_in[4];
    const float* w_ab = (const float*)d_in[5];
    const float* w_fc = (const float*)d_in[6];
    const float* w_fd = (const float*)d_in[7];
    const float* w_e  = (const float*)d_in[8];
    float* out = (float*)d_out;
    float* ws  = (float*)d_ws;

    // workspace layout (floats): 9*NPIX total = 14.75 MB
    float* avg = ws;
    float* mx  = ws + (size_t)NPIX;
    float* bm  = ws + 2*(size_t)NPIX;
    float* Ff  = ws + 3*(size_t)NPIX;      // 4 planes
    float* Cm  = ws + 7*(size_t)NPIX;
    float* Sm  = ws + 8*(size_t)NPIX;

    k1_reduce<<<(NPIX/4)/256, 256, 0, stream>>>(x, w_ab, avg, mx, bm);
    k2_convs<<<BATCH*100, 256, 0, stream>>>(avg, mx, w1, w3, w5, w7, w_fc, Ff, Cm);
    k3_attn<<<BATCH*HH, 160, 0, stream>>>(bm, Cm, Sm);
    dim3 g4(800, 4, BATCH);
    k4_gemm_we<<<g4, 256, 0, stream>>>(Ff, w_fd, w_e, out);   // G staged in d_out
    dim3 g5(CCH, BATCH);
    k5_sgemm_out<<<g5, 320, 0, stream>>>(Sm, x, out);         // in-place G -> out
}